// TreeTraversalTreeImpl_82884278878827
// MI455X (gfx1250) — compile-verified
//
#include <hip/hip_runtime.h>
#include <hip/hip_bf16.h>
#include <stdint.h>

#define BATCH      10000
#define NUM_TREES  1000
#define NUM_NODES  511
#define N_FEATURES 256
#define MAX_DEPTH  8

#define TILE_B  32
#define TILE_T  8
#define THREADS 256

// ---------------- CDNA5 async global -> LDS helpers ----------------
#if defined(__AMDGCN__) && __has_builtin(__builtin_amdgcn_global_load_async_to_lds_b128)
#define HAVE_ASYNC_LDS 1
#endif

// builtin signature (from hipcc diagnostic): (int4 AS1*, int4 AS3*, imm offset, imm cpol)
typedef int v4i __attribute__((ext_vector_type(4)));
typedef __attribute__((address_space(1))) v4i* gv4i_t;
typedef __attribute__((address_space(3))) v4i* lv4i_t;

__device__ __forceinline__ void copy_b128_to_lds(void* lds, const void* g) {
#ifdef HAVE_ASYNC_LDS
    __builtin_amdgcn_global_load_async_to_lds_b128((gv4i_t)g, (lv4i_t)lds, 0, 0);
#else
    *(float4*)lds = *(const float4*)g;   // synchronous fallback
#endif
}

__device__ __forceinline__ void wait_async_copies() {
#ifdef HAVE_ASYNC_LDS
#if __has_builtin(__builtin_amdgcn_s_wait_asynccnt)
    __builtin_amdgcn_s_wait_asynccnt(0);
#else
    asm volatile("s_wait_asynccnt 0" ::: "memory");
#endif
#endif
}

// ---------------- kernel ----------------
__global__ __launch_bounds__(THREADS) void TreeTraversal_kernel(
    const float* __restrict__ x,          // [BATCH, N_FEATURES]
    const float* __restrict__ thresholds, // [T*NN]
    const float* __restrict__ values,     // [T*NN, 1]
    const int*   __restrict__ lefts,      // [T*NN]
    const int*   __restrict__ rights,     // [T*NN]
    const int*   __restrict__ features,   // [T*NN]
    float*       __restrict__ out)        // [BATCH, NUM_TREES]
{
    __shared__ __align__(16) float s_x   [TILE_B * N_FEATURES];   // 32 KB
    __shared__ __align__(16) float s_thr [TILE_T * NUM_NODES];    // 16 KB
    __shared__ __align__(16) int   s_feat[TILE_T * NUM_NODES];
    __shared__ __align__(16) int   s_left[TILE_T * NUM_NODES];
    __shared__ __align__(16) int   s_rght[TILE_T * NUM_NODES];
    __shared__ __align__(16) float s_val [TILE_T * NUM_NODES];

    const int tid = threadIdx.x;
    const int b0  = blockIdx.x * TILE_B;          // batch tile base
    const int t0  = blockIdx.y * TILE_T;          // tree  tile base

    // ---- stage x tile (32 rows x 1KB), b128 granules, clamp tail rows ----
    #pragma unroll 2
    for (int i = tid; i < (TILE_B * N_FEATURES) / 4; i += THREADS) {
        int row    = i >> 6;                      // 64 float4 per row
        int srcRow = b0 + row;
        if (srcRow >= BATCH) srcRow = BATCH - 1;  // safe duplicate read
        const float* g = x + (size_t)srcRow * N_FEATURES + ((i & 63) << 2);
        copy_b128_to_lds(&s_x[i << 2], g);
    }

    // ---- stage 8 trees' node tables: each array is a contiguous 16,352B slice
    //      group base = t0*511 = 4088*g  -> float4 aligned (4088 % 4 == 0)  ----
    const size_t nbase = (size_t)t0 * NUM_NODES;
    for (int i = tid; i < (TILE_T * NUM_NODES) / 4; i += THREADS) {
        const int off = i << 2;
        copy_b128_to_lds(&s_thr [off], thresholds + nbase + off);
        copy_b128_to_lds(&s_feat[off], features   + nbase + off);
        copy_b128_to_lds(&s_left[off], lefts      + nbase + off);
        copy_b128_to_lds(&s_rght[off], rights     + nbase + off);
        copy_b128_to_lds(&s_val [off], values     + nbase + off);
    }

    wait_async_copies();   // drain this wave's ASYNCcnt before the barrier
    __syncthreads();       // make LDS visible workgroup-wide

    // ---- traversal: one (row, tree) pair per thread ----
    // lanes 0..7 of a wave share a row and cover 8 trees -> 32B-contiguous stores
    const int lt   = tid & (TILE_T - 1);   // local tree  0..7
    const int lb   = tid >> 3;             // local row   0..31
    const int base = lt * NUM_NODES;
    const float* xrow = &s_x[lb * N_FEATURES];

    int node = 0;
    #pragma unroll
    for (int d = 0; d < MAX_DEPTH; ++d) {
        const int   idx = base + node;
        const int   f   = s_feat[idx];
        const float th  = s_thr [idx];
        const int   l   = s_left[idx];
        const int   r   = s_rght[idx];
        const float xv  = xrow[f];
        node = (xv >= th) ? r : l;
    }

    const int b = b0 + lb;
    if (b < BATCH) {
        out[(size_t)b * NUM_TREES + (t0 + lt)] = s_val[base + node];
    }
}

// ---------------- launcher ----------------
extern "C" void kernel_launch(void* const* d_in, const int* in_sizes, int n_in,
                              void* d_out, int out_size, void* d_ws, size_t ws_size,
                              hipStream_t stream) {
    const float* x          = (const float*)d_in[0];
    const float* thresholds = (const float*)d_in[1];
    const float* values     = (const float*)d_in[2];
    const int*   lefts      = (const int*)  d_in[3];
    const int*   rights     = (const int*)  d_in[4];
    const int*   features   = (const int*)  d_in[5];
    // d_in[6..8]: num_trees / num_nodes / max_depth scalars (compile-time here)

    dim3 grid((BATCH + TILE_B - 1) / TILE_B, NUM_TREES / TILE_T);  // (313, 125)
    TreeTraversal_kernel<<<grid, dim3(THREADS), 0, stream>>>(
        x, thresholds, values, lefts, rights, features, (float*)d_out);
}